// TensorDLT_83528523972619
// MI455X (gfx1250) — compile-verified
//
#include <hip/hip_runtime.h>

typedef float v2f __attribute__((ext_vector_type(2)));
typedef float v8f __attribute__((ext_vector_type(8)));

#define NB 64  // batch size: 64 homographies

// One workgroup, 64 threads (2 wave32s). Lane s owns sample s.
// Phase 1: build A,b; Gauss-Jordan inverse with partial pivoting (registers only).
// Phase 2: stage A^-1 and per-tile stacked b to LDS; per wave, 16 block-diagonal
//          16x16 tiles (2 samples each) -> h = A^-1 b via chained
//          V_WMMA_F32_16X16X4_F32 with unconditional vectorized LDS reads.
__global__ __launch_bounds__(64) void tensordlt_kernel(
    const float* __restrict__ H4PT,
    const float* __restrict__ CornerA,
    float* __restrict__ out)
{
    __shared__ float sInv[NB][64];            // 16 KB: per-sample 8x8 inverse
    __shared__ float sBt[32][16];             //  2 KB: per-tile stacked [b(s0);b(s1)]
    __shared__ __align__(8) float zpad[20];   // zero pad for address-selected B reads

    const int s = threadIdx.x;  // sample index 0..63

    // ---------------- Phase 1: per-lane build + pivoted Gauss-Jordan ----------
    float M[8][8], Inv[8][8], bb[8];
    {
        float U[4], V[4], Ud[4], Vd[4];
        #pragma unroll
        for (int k = 0; k < 4; ++k) {
            float u = CornerA[s * 8 + 2 * k + 0];
            float v = CornerA[s * 8 + 2 * k + 1];
            U[k] = u; V[k] = v;
            Ud[k] = u + H4PT[s * 8 + 2 * k + 0];
            Vd[k] = v + H4PT[s * 8 + 2 * k + 1];
        }
        #pragma unroll
        for (int k = 0; k < 4; ++k) {
            // row 2k  (note +V, matching the reference's quirk)
            M[2*k][0] = 0.0f;  M[2*k][1] = 0.0f;  M[2*k][2] = 0.0f;
            M[2*k][3] = -U[k]; M[2*k][4] = V[k];  M[2*k][5] = -1.0f;
            M[2*k][6] = Vd[k] * U[k];  M[2*k][7] = Vd[k] * V[k];
            bb[2*k] = -Vd[k];
            // row 2k+1
            M[2*k+1][0] = U[k]; M[2*k+1][1] = V[k]; M[2*k+1][2] = 1.0f;
            M[2*k+1][3] = 0.0f; M[2*k+1][4] = 0.0f; M[2*k+1][5] = 0.0f;
            M[2*k+1][6] = -Ud[k] * U[k];  M[2*k+1][7] = -Ud[k] * V[k];
            bb[2*k+1] = Ud[k];
        }
        #pragma unroll
        for (int i = 0; i < 8; ++i)
            #pragma unroll
            for (int j = 0; j < 8; ++j)
                Inv[i][j] = (i == j) ? 1.0f : 0.0f;

        #pragma unroll
        for (int k = 0; k < 8; ++k) {
            // partial pivoting: branch-free conditional row swaps (constant idx)
            #pragma unroll
            for (int j = k + 1; j < 8; ++j) {
                bool sw = __builtin_fabsf(M[j][k]) > __builtin_fabsf(M[k][k]);
                #pragma unroll
                for (int c = 0; c < 8; ++c) {
                    float mk = M[k][c],  mj = M[j][c];
                    M[k][c]   = sw ? mj : mk;   M[j][c]   = sw ? mk : mj;
                    float ik = Inv[k][c], ij = Inv[j][c];
                    Inv[k][c] = sw ? ij : ik;   Inv[j][c] = sw ? ik : ij;
                }
            }
            float piv = 1.0f / M[k][k];
            #pragma unroll
            for (int c = 0; c < 8; ++c) { M[k][c] *= piv; Inv[k][c] *= piv; }
            #pragma unroll
            for (int j = 0; j < 8; ++j) {
                if (j == k) continue;
                float f = M[j][k];
                #pragma unroll
                for (int c = 0; c < 8; ++c) {
                    M[j][c]   -= f * M[k][c];
                    Inv[j][c] -= f * Inv[k][c];
                }
            }
        }
    }

    // ---------------- stage to LDS ---------------------------------------------
    #pragma unroll
    for (int i = 0; i < 8; ++i)
        #pragma unroll
        for (int j = 0; j < 8; ++j)
            sInv[s][i * 8 + j] = Inv[i][j];
    // stacked b per tile: tile t = s>>1 holds [b(2t) ; b(2t+1)]
    #pragma unroll
    for (int i = 0; i < 8; ++i) sBt[s >> 1][(s & 1) * 8 + i] = bb[i];
    if (s < 20) zpad[s] = 0.0f;

    __syncthreads();

    // ---------------- Phase 2: block-diagonal WMMA matvec ----------------------
    const int wave  = threadIdx.x >> 5;
    const int lane  = threadIdx.x & 31;
    const int m     = lane & 15;   // tile row owned by this lane (A frag)
    const int khalf = lane >> 4;   // 0: K pair {0,1}; 1: K pair {2,3} of each slab
    const bool mlow = (m < 8);
    const v2f  z2   = {0.0f, 0.0f};

    for (int t = 0; t < 16; ++t) {
        const int s0 = wave * 32 + 2 * t;
        const int s1 = s0 + 1;
        const int gt = wave * 16 + t;   // global tile index for sBt

        // Unconditional vectorized operand fetch: EXEC untouched.
        // A: lane's row of its own sample's inverse (always a valid address).
        const float* abase = mlow ? &sInv[s0][m * 8] : &sInv[s1][(m - 8) * 8];
        const v2f aR0 = *(const v2f*)(abase + 2 * khalf);       // cols {0,1}|{2,3}
        const v2f aR1 = *(const v2f*)(abase + 2 * khalf + 4);   // cols {4,5}|{6,7}
        // Block-diagonal: lanes m<8 contribute on K-slabs 0,1; m>=8 on slabs 2,3.
        const v2f a0 = mlow ? aR0 : z2;
        const v2f a1 = mlow ? aR1 : z2;
        const v2f a2 = mlow ? z2  : aR0;
        const v2f a3 = mlow ? z2  : aR1;

        // B: column N==0 holds the stacked b; lanes with N!=0 read the zero pad.
        const float* bbase = (m == 0) ? &sBt[gt][2 * khalf] : &zpad[2 * khalf];
        const v2f b0 = *(const v2f*)(bbase + 0);
        const v2f b1 = *(const v2f*)(bbase + 4);
        const v2f b2 = *(const v2f*)(bbase + 8);
        const v2f b3 = *(const v2f*)(bbase + 12);

        // D = diag(Inv0, Inv1) x Bstacked, chained over K = 16.
        v8f acc;
        acc = __builtin_amdgcn_wmma_f32_16x16x4_f32(
            false, a0, false, b0, (short)0, v8f{0,0,0,0,0,0,0,0}, false, false);
        acc = __builtin_amdgcn_wmma_f32_16x16x4_f32(
            false, a1, false, b1, (short)0, acc, false, false);
        acc = __builtin_amdgcn_wmma_f32_16x16x4_f32(
            false, a2, false, b2, (short)0, acc, false, false);
        acc = __builtin_amdgcn_wmma_f32_16x16x4_f32(
            false, a3, false, b3, (short)0, acc, false, false);

        // Column N=0 of C: lane 0 holds M=0..7 = h(s0); lane 16 holds M=8..15 = h(s1)
        if (m == 0) {
            const int so = (khalf == 0) ? s0 : s1;
            #pragma unroll
            for (int r = 0; r < 8; ++r) out[so * 9 + r] = acc[r];
            out[so * 9 + 8] = 1.0f;
        }
    }
}

extern "C" void kernel_launch(void* const* d_in, const int* in_sizes, int n_in,
                              void* d_out, int out_size, void* d_ws, size_t ws_size,
                              hipStream_t stream) {
    (void)in_sizes; (void)n_in; (void)out_size; (void)d_ws; (void)ws_size;
    const float* H4PT    = (const float*)d_in[0];
    const float* CornerA = (const float*)d_in[1];
    float* out = (float*)d_out;
    // Single workgroup: 64 lanes = 64 samples, 2 wave32s; launch-latency bound.
    tensordlt_kernel<<<1, 64, 0, stream>>>(H4PT, CornerA, out);
}